// DangoPreTrain_51900384805106
// MI455X (gfx1250) — compile-verified
//
#include <hip/hip_runtime.h>
#include <hip/hip_bf16.h>

// ---------------- problem constants (from reference) ----------------
constexpr int N = 6607;        // gene_num
constexpr int H = 64;          // hidden channels
constexpr int T = 3;           // edge types
constexpr int E = 1000000;     // edges per type

typedef float v2f __attribute__((ext_vector_type(2)));
typedef float v8f __attribute__((ext_vector_type(8)));

// ---------------- utility kernels ----------------
__global__ void zero_f32(float* __restrict__ p, size_t n) {
    size_t i = (size_t)blockIdx.x * blockDim.x + threadIdx.x;
    size_t stride = (size_t)gridDim.x * blockDim.x;
    for (; i < n; i += stride) p[i] = 0.0f;
}

__global__ void copy_f32(const float* __restrict__ src, float* __restrict__ dst, size_t n) {
    size_t i = (size_t)blockIdx.x * blockDim.x + threadIdx.x;
    size_t stride = (size_t)gridDim.x * blockDim.x;
    for (; i < n; i += stride) dst[i] = src[i];
}

// ---------------- degree: deg[t][dst] += 1 ----------------
__global__ void degree_kernel(const int* __restrict__ edge_index, float* __restrict__ deg) {
    long total = (long)T * E;
    long stride = (long)gridDim.x * blockDim.x;
    for (long idx = (long)blockIdx.x * blockDim.x + threadIdx.x; idx < total; idx += stride) {
        int t = (int)(idx / E);
        int e = (int)(idx - (long)t * E);
        int d = edge_index[(size_t)t * 2 * E + E + e];
        atomicAdd(&deg[t * N + d], 1.0f);
    }
}

// ---------------- scatter: agg[t][dst][h] += x[(t)][src][h] ----------------
// xTypeStride = 0 for shared x (layer 1), N*H for per-type x (layer 2)
__global__ void scatter_kernel(const float* __restrict__ x, int xTypeStride,
                               const int* __restrict__ edge_index,
                               float* __restrict__ agg) {
    int t = blockIdx.z;
    const int* srcp = edge_index + (size_t)t * 2 * E;
    const int* dstp = srcp + E;
    const float* xt = x + (size_t)t * xTypeStride;
    float* aggt = agg + (size_t)t * N * H;
    long total = (long)E * H;
    long stride = (long)gridDim.x * blockDim.x;
    for (long idx = (long)blockIdx.x * blockDim.x + threadIdx.x; idx < total; idx += stride) {
        int e  = (int)(idx >> 6);   // H == 64
        int hh = (int)(idx & 63);
        int s = srcp[e];
        int d = dstp[e];
        float v = xt[(size_t)s * H + hh];
        atomicAdd(&aggt[(size_t)d * H + hh], v);
    }
}

// ---------------- SAGE dense: out = relu(mean_agg @ Wl^T + b + x @ Wr^T) ----------------
// block = (64, 4): x = output channel h, y = row within block
__global__ void sage_dense_kernel(const float* __restrict__ x, int xTypeStride,
                                  const float* __restrict__ agg,
                                  const float* __restrict__ deg,
                                  const float* __restrict__ Wl,
                                  const float* __restrict__ bias,
                                  const float* __restrict__ Wr,
                                  float* __restrict__ out) {
    __shared__ float wl[H * H];
    __shared__ float wr[H * H];
    __shared__ float xs[4][H];
    __shared__ float as[4][H];

    int t = blockIdx.z;
    int tid = threadIdx.y * 64 + threadIdx.x;
    for (int i = tid; i < H * H; i += 256) {
        wl[i] = Wl[(size_t)t * H * H + i];
        wr[i] = Wr[(size_t)t * H * H + i];
    }

    int r  = blockIdx.x * 4 + threadIdx.y;
    int rr = (r < N) ? r : (N - 1);
    xs[threadIdx.y][threadIdx.x] = x[(size_t)t * xTypeStride + (size_t)rr * H + threadIdx.x];
    float dg  = deg[t * N + rr];
    float inv = 1.0f / fmaxf(dg, 1.0f);
    as[threadIdx.y][threadIdx.x] = agg[((size_t)t * N + rr) * H + threadIdx.x] * inv;
    __syncthreads();

    if (r >= N) return;
    int h = threadIdx.x;
    float acc = bias[t * H + h];
#pragma unroll 8
    for (int k = 0; k < H; ++k) {
        acc += as[threadIdx.y][k] * wl[h * H + k];
        acc += xs[threadIdx.y][k] * wr[h * H + k];
    }
    out[((size_t)t * N + r) * H + h] = fmaxf(acc, 0.0f);
}

// ---------------- recon: out[t] = h2[t] @ Wrec[t]^T + brec[t]  (WMMA fp32) ----------------
// One wave computes a 32x32 output tile as 2x2 V_WMMA_F32_16X16X4_F32 subtiles,
// looping K = 0..63 in steps of 4. A/B VGPR layouts per ISA 7.12.2:
//   lane l: half = l>>4, l15 = l&15
//   A (16x4):  a[v] = A[M0 + 16*si + l15][k0 + 2*half + v]   -> contiguous float2
//   B (4x16):  b[v] = B[k0 + 2*half + v][N0 + 16*sj + l15] = Wrec[col][k] -> contiguous float2
//   D (16x16): d[r] = C[16*si + r + 8*half][16*sj + l15]
constexpr int TILES_1D    = (N + 31) / 32;          // 207
constexpr int TILES_TOTAL = TILES_1D * TILES_1D;    // 42849 per type
constexpr int WAVES_PER_BLOCK = 8;

__global__ void recon_wmma_kernel(const float* __restrict__ h2,    // [T,N,H]
                                  const float* __restrict__ Wrec,  // [T,N,H]
                                  const float* __restrict__ brec,  // [T,N]
                                  float* __restrict__ out) {       // [T,N,N]
    int t    = blockIdx.y;
    int wave = threadIdx.x >> 5;
    int lane = threadIdx.x & 31;
    int tile = blockIdx.x * WAVES_PER_BLOCK + wave;
    if (tile >= TILES_TOTAL) return;          // wave-uniform: EXEC stays all-ones

    int tm = tile / TILES_1D;
    int tn = tile - tm * TILES_1D;
    int M0 = tm * 32;
    int N0 = tn * 32;

    const float* A = h2   + (size_t)t * N * H;
    const float* B = Wrec + (size_t)t * N * H;

    int half = lane >> 4;
    int l15  = lane & 15;

    // clamped rows (branchless; padding rows produce garbage that is never stored)
    int ra0 = min(M0 + l15,      N - 1);
    int ra1 = min(M0 + 16 + l15, N - 1);
    int rb0 = min(N0 + l15,      N - 1);
    int rb1 = min(N0 + 16 + l15, N - 1);

    const float* pa0 = A + (size_t)ra0 * H + 2 * half;
    const float* pa1 = A + (size_t)ra1 * H + 2 * half;
    const float* pb0 = B + (size_t)rb0 * H + 2 * half;
    const float* pb1 = B + (size_t)rb1 * H + 2 * half;

    v8f c00 = {}, c01 = {}, c10 = {}, c11 = {};
#pragma unroll
    for (int k0 = 0; k0 < H; k0 += 4) {
        v2f a0 = *(const v2f*)(pa0 + k0);
        v2f a1 = *(const v2f*)(pa1 + k0);
        v2f b0 = *(const v2f*)(pb0 + k0);
        v2f b1 = *(const v2f*)(pb1 + k0);
        c00 = __builtin_amdgcn_wmma_f32_16x16x4_f32(false, a0, false, b0, (short)0, c00, false, false);
        c01 = __builtin_amdgcn_wmma_f32_16x16x4_f32(false, a0, false, b1, (short)0, c01, false, false);
        c10 = __builtin_amdgcn_wmma_f32_16x16x4_f32(false, a1, false, b0, (short)0, c10, false, false);
        c11 = __builtin_amdgcn_wmma_f32_16x16x4_f32(false, a1, false, b1, (short)0, c11, false, false);
    }

    // epilogue: add brec[col], bounds-checked non-temporal stores (keep L2 for A/B)
    const float* bb = brec + (size_t)t * N;
    int c0 = N0 + l15;
    int c1 = N0 + 16 + l15;
    float bias0 = bb[min(c0, N - 1)];
    float bias1 = bb[min(c1, N - 1)];
    float* O = out + (size_t)t * N * N;

#pragma unroll
    for (int r = 0; r < 8; ++r) {
        int row0 = M0 + r + 8 * half;        // si = 0
        int row1 = M0 + 16 + r + 8 * half;   // si = 1
        if (row0 < N) {
            if (c0 < N) __builtin_nontemporal_store(c00[r] + bias0, O + (size_t)row0 * N + c0);
            if (c1 < N) __builtin_nontemporal_store(c01[r] + bias1, O + (size_t)row0 * N + c1);
        }
        if (row1 < N) {
            if (c0 < N) __builtin_nontemporal_store(c10[r] + bias0, O + (size_t)row1 * N + c0);
            if (c1 < N) __builtin_nontemporal_store(c11[r] + bias1, O + (size_t)row1 * N + c1);
        }
    }
}

// ---------------- launch ----------------
extern "C" void kernel_launch(void* const* d_in, const int* in_sizes, int n_in,
                              void* d_out, int out_size, void* d_ws, size_t ws_size,
                              hipStream_t stream) {
    const float* emb        = (const float*)d_in[0];   // [N,H]
    const int*   edge_index = (const int*)  d_in[1];   // [T,2,E]
    const float* W1l        = (const float*)d_in[2];   // [T,H,H]
    const float* b1         = (const float*)d_in[3];   // [T,H]
    const float* W1r        = (const float*)d_in[4];   // [T,H,H]
    const float* W2l        = (const float*)d_in[5];   // [T,H,H]
    const float* b2         = (const float*)d_in[6];   // [T,H]
    const float* W2r        = (const float*)d_in[7];   // [T,H,H]
    const float* Wrec       = (const float*)d_in[8];   // [T,N,H]
    const float* brec       = (const float*)d_in[9];   // [T,N]

    // output layout: embeddings [T,N,H] | reconstructions [T,N,N] | x_init [N,H]
    float* out       = (float*)d_out;
    float* emb_out   = out;
    float* recon_out = out + (size_t)T * N * H;
    float* xinit_out = recon_out + (size_t)T * N * N;

    // workspace layout (floats): deg [T,N] | agg [T,N,H] | h1 [T,N,H]
    float* ws  = (float*)d_ws;
    float* deg = ws;
    float* agg = ws + 32768;                       // padded offset
    float* h1  = agg + (size_t)T * N * H;

    const size_t nDeg = (size_t)T * N;
    const size_t nAgg = (size_t)T * N * H;

    dim3 blk256(256);
    dim3 scatterGrid(4096, 1, T);
    dim3 denseGrid((N + 3) / 4, 1, T);
    dim3 denseBlk(64, 4);
    dim3 reconGrid((TILES_TOTAL + WAVES_PER_BLOCK - 1) / WAVES_PER_BLOCK, T);

    // ---- layer 1 ----
    zero_f32<<<256, blk256, 0, stream>>>(deg, nDeg);
    zero_f32<<<2048, blk256, 0, stream>>>(agg, nAgg);
    degree_kernel<<<2048, blk256, 0, stream>>>(edge_index, deg);
    scatter_kernel<<<scatterGrid, blk256, 0, stream>>>(emb, 0, edge_index, agg);
    sage_dense_kernel<<<denseGrid, denseBlk, 0, stream>>>(emb, 0, agg, deg, W1l, b1, W1r, h1);

    // ---- layer 2 ----
    zero_f32<<<2048, blk256, 0, stream>>>(agg, nAgg);
    scatter_kernel<<<scatterGrid, blk256, 0, stream>>>(h1, N * H, edge_index, agg);
    sage_dense_kernel<<<denseGrid, denseBlk, 0, stream>>>(h1, N * H, agg, deg, W2l, b2, W2r, emb_out);

    // ---- x_init passthrough ----
    copy_f32<<<512, blk256, 0, stream>>>(emb, xinit_out, (size_t)N * H);

    // ---- reconstruction GEMM (WMMA fp32) ----
    recon_wmma_kernel<<<reconGrid, blk256, 0, stream>>>(emb_out, Wrec, brec, recon_out);
}